// QuantumClassifier_16475494547731
// MI455X (gfx1250) — compile-verified
//
#include <hip/hip_runtime.h>
#include <hip/hip_bf16.h>

typedef __attribute__((ext_vector_type(2))) float v2f;
typedef __attribute__((ext_vector_type(8))) float v8f;

#define HALF_PI 1.57079632679489662f
#define TILES_PER_WAVE 4

// ---------------------------------------------------------------------------
// Setup kernel: build A = Re(U^dag Z0 U) from the 16 weights, where U is the
// batch-independent part of the circuit (2 layers of RY/RZ + CNOT ring).
// Store A pre-swizzled into ws in the exact per-lane order the main kernel's
// WMMA A-operand expects (32-bit A 16x4 layout, ISA 7.12.2).
// ---------------------------------------------------------------------------
__global__ __launch_bounds__(32) void qc_setup(const float* __restrict__ w,
                                               float* __restrict__ wsA) {
    __shared__ float Ur[256];   // Ur[t*16 + k] = Re(column t of U)[k]
    __shared__ float Ui[256];
    __shared__ float Amat[256]; // Amat[i*16 + j] = A[i][j]

    const int t = threadIdx.x;

    if (t < 16) {
        // state = basis vector e_t (qubit0 = bit 8, qubit3 = bit 1)
        float re[16], im[16];
#pragma unroll
        for (int i = 0; i < 16; ++i) { re[i] = (i == t) ? 1.f : 0.f; im[i] = 0.f; }

#pragma unroll
        for (int layer = 0; layer < 2; ++layer) {
            const int base = layer * 8;
#pragma unroll
            for (int q = 0; q < 4; ++q) {
                const int m = 8 >> q;
                // RY(w[base+q]) on qubit q
                float sh, ch;
                __sincosf(0.5f * w[base + q], &sh, &ch);
#pragma unroll
                for (int i = 0; i < 16; ++i) {
                    if (!(i & m)) {
                        const int i1 = i | m;
                        const float r0 = re[i], m0 = im[i];
                        const float r1 = re[i1], m1 = im[i1];
                        re[i]  = ch * r0 - sh * r1;  im[i]  = ch * m0 - sh * m1;
                        re[i1] = sh * r0 + ch * r1;  im[i1] = sh * m0 + ch * m1;
                    }
                }
                // RZ(w[base+4+q]) on qubit q: bit0 *= e^{-i h}, bit1 *= e^{+i h}
                float shz, chz;
                __sincosf(0.5f * w[base + 4 + q], &shz, &chz);
#pragma unroll
                for (int i = 0; i < 16; ++i) {
                    const float r = re[i], mm = im[i];
                    if (!(i & m)) { re[i] = r * chz + mm * shz; im[i] = mm * chz - r * shz; }
                    else          { re[i] = r * chz - mm * shz; im[i] = mm * chz + r * shz; }
                }
            }
            // CNOT chain (q -> q+1) then ring closure (q3 -> q0)
#pragma unroll
            for (int q = 0; q < 3; ++q) {
                const int cm = 8 >> q, tm = 8 >> (q + 1);
#pragma unroll
                for (int i = 0; i < 16; ++i) {
                    if ((i & cm) && !(i & tm)) {
                        const int i1 = i | tm;
                        float r = re[i], mm = im[i];
                        re[i] = re[i1]; im[i] = im[i1];
                        re[i1] = r;     im[i1] = mm;
                    }
                }
            }
            {
                const int cm = 1, tm = 8; // control qubit3, target qubit0
#pragma unroll
                for (int i = 0; i < 16; ++i) {
                    if ((i & cm) && !(i & tm)) {
                        const int i1 = i | tm;
                        float r = re[i], mm = im[i];
                        re[i] = re[i1]; im[i] = im[i1];
                        re[i1] = r;     im[i1] = mm;
                    }
                }
            }
        }
#pragma unroll
        for (int k = 0; k < 16; ++k) { Ur[t * 16 + k] = re[k]; Ui[t * 16 + k] = im[k]; }
    }
    __syncthreads();

    if (t < 16) {
        // A[i][t] = sum_k sign(k) * Re( conj(U[k][i]) * U[k][t] ), sign = +1 if k<8
#pragma unroll
        for (int i = 0; i < 16; ++i) {
            float acc = 0.f;
#pragma unroll
            for (int k = 0; k < 16; ++k) {
                const float sgn = (k & 8) ? -1.f : 1.f;
                acc += sgn * (Ur[i * 16 + k] * Ur[t * 16 + k] +
                              Ui[i * 16 + k] * Ui[t * 16 + k]);
            }
            Amat[i * 16 + t] = acc;
        }
    }
    __syncthreads();

    // Pre-swizzle for WMMA A operand: wsA[(c*32 + lane)*2 + v] = A[m][k]
    // with m = lane&15, k = 4c + 2*(lane>>4) + v   (c = K-chunk index 0..3)
    for (int p = t; p < 256; p += 32) {
        const int lane = (p >> 1) & 31;
        const int v    = p & 1;
        const int c    = p >> 6;
        const int m    = lane & 15;
        const int k    = 4 * c + 2 * (lane >> 4) + v;
        wsA[p] = Amat[m * 16 + k];
    }
}

// sin(pi*x/2) & cos(pi*x/2) via raw hardware ops (input in revolutions):
// sin(2*pi*(x/4)) -> one mul instead of __sincosf's two-step reduction.
__device__ __forceinline__ void sincos_half_pi(float x, float& s, float& c) {
    const float r = 0.25f * x;
    s = __builtin_amdgcn_sinf(r);
    c = __builtin_amdgcn_cosf(r);
}

// One 16-element tile: build factors, 4 chained WMMAs, factored column dot,
// lane-pair sum via ds_swizzle SWAPX16. Returns per-lane total (valid for
// lanes with h==0; h==1 lanes hold the same value after the swizzle-add).
__device__ __forceinline__ float qc_tile(const float4 xv, const v2f a_op[4], int h) {
    float s0, c0, s1, c1, s2, c2, s3, c3;
    sincos_half_pi(xv.x, s0, c0);
    sincos_half_pi(xv.y, s1, c1);
    sincos_half_pi(xv.z, s2, c2);
    sincos_half_pi(xv.w, s3, c3);

    const float ab0 = c0 * c1, ab1 = c0 * s1, ab2 = s0 * c1, ab3 = s0 * s1;
    const float cd0 = c2 * c3, cd1 = c2 * s3, cd2 = s2 * c3, cd3 = s2 * s3;

    // Half-wave-dependent factors (4 cndmasks per tile).
    const float cdh0 = h ? cd2 : cd0;
    const float cdh1 = h ? cd3 : cd1;
    const float abh0 = h ? ab2 : ab0;
    const float abh1 = h ? ab3 : ab1;

    const float abv[4] = {ab0, ab1, ab2, ab3};

    // D = A * S via 4 chained f32 WMMAs (K = 4 each).
    // B-operand chunk c: S[4c+2h+v][j] = ab[c] * cd[2h+v].
    v8f acc = {};
#pragma unroll
    for (int c = 0; c < 4; ++c) {
        const v2f b_op = (v2f){abv[c] * cdh0, abv[c] * cdh1};
        acc = __builtin_amdgcn_wmma_f32_16x16x4_f32(
            false, a_op[c], false, b_op, (short)0, acc, false, false);
    }

    // out_j = sum_i S[i][j]*D[i][j]; rows 8h..8h+7 live in this lane,
    // S[8h+r][j] = ab[2h+(r>>2)] * cd[r&3]  ->  factor through cd.
    float d0 = cd0 * acc[0];
    d0 = fmaf(cd1, acc[1], d0);
    d0 = fmaf(cd2, acc[2], d0);
    d0 = fmaf(cd3, acc[3], d0);
    float d1 = cd0 * acc[4];
    d1 = fmaf(cd1, acc[5], d1);
    d1 = fmaf(cd2, acc[6], d1);
    d1 = fmaf(cd3, acc[7], d1);
    const float partial = abh0 * d0 + abh1 * d1;

    // Lane-pair sum (j <-> j+16): ds_swizzle SWAPX16 (xor=0x10, and=0x1f).
    const int pi = __builtin_bit_cast(int, partial);
    const int qi = __builtin_amdgcn_ds_swizzle(pi, 0x401f);
    return partial + __builtin_bit_cast(float, qi);
}

__global__ __launch_bounds__(256) void qc_main(const float* __restrict__ x,
                                               const float* __restrict__ wsA,
                                               float* __restrict__ out,
                                               int nB) {
    const int tid  = blockIdx.x * blockDim.x + threadIdx.x;
    const int lane = threadIdx.x & 31;
    const int wave = tid >> 5;
    const int j    = lane & 15;   // column (batch element within tile)
    const int h    = lane >> 4;   // half-wave selector

    // Load constant A operand chunks (1 KB total, hot in WGP$/L2).
    v2f a_op[4];
#pragma unroll
    for (int c = 0; c < 4; ++c) {
        const float2 t2 = *reinterpret_cast<const float2*>(wsA + (c * 32 + lane) * 2);
        a_op[c] = (v2f){t2.x, t2.y};
    }

    const long base = (long)wave * (16 * TILES_PER_WAVE);

    if (base + 16 * TILES_PER_WAVE <= (long)nB) {
        // ------- fast path: no bounds checks, affine immediate-offset access
        const float4* xp = reinterpret_cast<const float4*>(x + 4 * (base + j));
        float totals[TILES_PER_WAVE];
#pragma unroll
        for (int t = 0; t < TILES_PER_WAVE; ++t)
            totals[t] = qc_tile(xp[t * 16], a_op, h);
        if (h == 0) {
            float* op = out + base + j;
#pragma unroll
            for (int t = 0; t < TILES_PER_WAVE; ++t)
                op[t * 16] = totals[t];
        }
    } else {
        // ------- tail path (only if nB % 64 != 0)
#pragma unroll
        for (int t = 0; t < TILES_PER_WAVE; ++t) {
            const long eb = base + (long)t * 16;
            if (eb >= nB) break;            // wave-uniform
            long e = eb + j;
            if (e >= nB) e = nB - 1;        // clamp (store is guarded)
            const float4 xv = *reinterpret_cast<const float4*>(x + 4 * e);
            const float total = qc_tile(xv, a_op, h);
            if (h == 0 && (eb + j) < nB) out[eb + j] = total;
        }
    }
}

extern "C" void kernel_launch(void* const* d_in, const int* in_sizes, int n_in,
                              void* d_out, int out_size, void* d_ws, size_t ws_size,
                              hipStream_t stream) {
    const float* x = (const float*)d_in[0];   // [B,4] float32
    const float* w = (const float*)d_in[1];   // [16]  float32
    float* out = (float*)d_out;               // [B]   float32
    float* wsA = (float*)d_ws;                // 256 floats scratch

    const int nB = in_sizes[0] / 4;

    qc_setup<<<dim3(1), dim3(32), 0, stream>>>(w, wsA);

    const long waves   = ((long)nB + 16 * TILES_PER_WAVE - 1) / (16 * TILES_PER_WAVE);
    const long threads = waves * 32;
    const int  block   = 256;
    const long grid    = (threads + block - 1) / block;
    qc_main<<<dim3((unsigned)grid), dim3(block), 0, stream>>>(x, wsA, out, nB);
}